// RgcnConv_3728031613523
// MI455X (gfx1250) — compile-verified
//
#include <hip/hip_runtime.h>

#define N_NODES   100000
#define N_EDGES   1600000
#define IN_FEAT   128
#define OUT_FEAT  128
#define NUM_RELS  8
#define NUM_BASES 2
#define EDGES_PER_WAVE 256

// GEMM tiling
#define KC         32              // K-chunk staged in LDS
#define LDS_STRIDE (KC + 2)        // pad to spread LDS banks, keeps 8B align
#define MT_PER_BLK 8               // 8 M-tiles of 16 rows -> 128 rows/block

typedef __attribute__((ext_vector_type(2))) float v2f;
typedef __attribute__((ext_vector_type(8))) float v8f;

// ---------------------------------------------------------------------------
// Kernel 0: zero the aggregation scratch (harness poisons ws with 0xAA)
// ---------------------------------------------------------------------------
__global__ void rgcn_zero_ws(float4* __restrict__ p, int n4) {
    int i = blockIdx.x * blockDim.x + threadIdx.x;
    if (i < n4) p[i] = float4{0.f, 0.f, 0.f, 0.f};
}

// ---------------------------------------------------------------------------
// Kernel 1: Wcomb = [ W0 + loop_weight | W1 | W2 ]  (384 x 128)
// ---------------------------------------------------------------------------
__global__ void rgcn_build_wc(const float* __restrict__ W,
                              const float* __restrict__ loopw,
                              float* __restrict__ wc) {
    int i = blockIdx.x * blockDim.x + threadIdx.x;
    const int total = (NUM_BASES + 1) * IN_FEAT * OUT_FEAT;
    if (i < total) {
        float v = W[i];
        if (i < IN_FEAT * OUT_FEAT) v += loopw[i];
        wc[i] = v;
    }
}

// ---------------------------------------------------------------------------
// Kernel 2: edge aggregation. One wave owns EDGES_PER_WAVE consecutive edges.
// edge_dst is sorted -> accumulate same-dst runs in registers, flush with
// global_atomic_add_f32 only on dst change. Lane l owns features [4l, 4l+4).
// ---------------------------------------------------------------------------
__device__ __forceinline__ void rgcn_flush(float* __restrict__ agg, int dst, int lane,
                                           float4& a0, float4& a1) {
    float* p = agg + (size_t)dst * (2 * IN_FEAT) + lane * 4;
    unsafeAtomicAdd(p + 0, a0.x);
    unsafeAtomicAdd(p + 1, a0.y);
    unsafeAtomicAdd(p + 2, a0.z);
    unsafeAtomicAdd(p + 3, a0.w);
    float* q = p + IN_FEAT;
    unsafeAtomicAdd(q + 0, a1.x);
    unsafeAtomicAdd(q + 1, a1.y);
    unsafeAtomicAdd(q + 2, a1.z);
    unsafeAtomicAdd(q + 3, a1.w);
    a0 = float4{0.f, 0.f, 0.f, 0.f};
    a1 = float4{0.f, 0.f, 0.f, 0.f};
}

__global__ void __launch_bounds__(256) rgcn_edge_agg(
        const float* __restrict__ feat, const float* __restrict__ coeff,
        const int* __restrict__ esrc, const int* __restrict__ edst,
        const int* __restrict__ etyp, float* __restrict__ agg) {
    const int wave = (blockIdx.x * blockDim.x + threadIdx.x) >> 5;
    const int lane = threadIdx.x & 31;
    long long e0 = (long long)wave * EDGES_PER_WAVE;
    if (e0 >= N_EDGES) return;
    long long e1 = e0 + EDGES_PER_WAVE;
    if (e1 > N_EDGES) e1 = N_EDGES;

    float4 acc0{0.f, 0.f, 0.f, 0.f};
    float4 acc1{0.f, 0.f, 0.f, 0.f};
    int dcur = edst[e0];

    for (long long base = e0; base < e1; base += 32) {
        // batch-load 32 edges' indices, one per lane, broadcast via shuffles
        int s = 0, d = dcur, t = 0;
        long long eL = base + lane;
        if (eL < e1) { s = esrc[eL]; d = edst[eL]; t = etyp[eL]; }
        // hint: pull the gathered rows toward the caches (global_prefetch_b8)
        __builtin_prefetch(feat + (size_t)s * IN_FEAT, 0, 3);

        int nb = (int)(e1 - base); if (nb > 32) nb = 32;
        for (int i = 0; i < nb; ++i) {
            int si = __shfl(s, i);
            int di = __shfl(d, i);
            int ti = __shfl(t, i);
            if (di != dcur) {              // uniform across wave (broadcast)
                rgcn_flush(agg, dcur, lane, acc0, acc1);
                dcur = di;
            }
            const float4 m = *(const float4*)(feat + (size_t)si * IN_FEAT + lane * 4);
            const float w0 = coeff[ti * NUM_BASES + 0];
            const float w1 = coeff[ti * NUM_BASES + 1];
            acc0.x += w0 * m.x; acc0.y += w0 * m.y; acc0.z += w0 * m.z; acc0.w += w0 * m.w;
            acc1.x += w1 * m.x; acc1.y += w1 * m.y; acc1.z += w1 * m.z; acc1.w += w1 * m.w;
        }
    }
    rgcn_flush(agg, dcur, lane, acc0, acc1);
}

// ---------------------------------------------------------------------------
// Kernel 3: h = [feat | agg0 | agg1] @ Wcomb + bias via V_WMMA_F32_16X16X4_F32
// Block = 256 thr (8 waves) computes a 128x128 output tile. The 128-row A
// stripe is staged through LDS per 32-wide K-chunk (each A element fetched
// from global once per block); each wave owns one 16-col N-tile and 8
// accumulators, reusing each B fragment across all 8 M-tiles.
// ---------------------------------------------------------------------------
__global__ void __launch_bounds__(256) rgcn_gemm(
        const float* __restrict__ feat, const float* __restrict__ agg,
        const float* __restrict__ wc, const float* __restrict__ bias,
        float* __restrict__ out) {
    __shared__ float As[128 * LDS_STRIDE];            // 17 KB

    const int lane  = threadIdx.x & 31;
    const int nt    = threadIdx.x >> 5;               // N tile (0..7)
    const int mbase = blockIdx.x * 128;               // block row base
    const int koff  = (lane >> 4) * 2;                // K sub-offset: 0 or 2
    const int ncol  = nt * 16 + (lane & 15);          // B/D column

    v8f acc[MT_PER_BLK];
#pragma unroll
    for (int mt = 0; mt < MT_PER_BLK; ++mt)
        acc[mt] = v8f{0.f, 0.f, 0.f, 0.f, 0.f, 0.f, 0.f, 0.f};

    for (int seg = 0; seg < NUM_BASES + 1; ++seg) {
        for (int k0 = 0; k0 < IN_FEAT; k0 += KC) {
            __syncthreads();                          // protect prior reads
            // ---- stage A[128][KC] chunk into LDS (one float4 per thread x4)
#pragma unroll
            for (int it = 0; it < 4; ++it) {
                int idx = it * 256 + threadIdx.x;     // 0..1023
                int row = idx >> 3;                   // KC/4 == 8 quads/row
                int c4  = (idx & 7) << 2;
                int grow = mbase + row;
                float4 v{0.f, 0.f, 0.f, 0.f};
                if (grow < N_NODES) {
                    const float* src = (seg == 0)
                        ? feat + (size_t)grow * IN_FEAT + k0 + c4
                        : agg + (size_t)grow * (2 * IN_FEAT) + (seg - 1) * IN_FEAT + k0 + c4;
                    v = *(const float4*)src;
                }
                float* dp = &As[row * LDS_STRIDE + c4];
                dp[0] = v.x; dp[1] = v.y; dp[2] = v.z; dp[3] = v.w;
            }
            __syncthreads();

            // ---- MACs: B frag loaded once per k-step, reused over 8 M-tiles
            const float* B = wc + seg * IN_FEAT * OUT_FEAT + ncol;
#pragma unroll
            for (int k = 0; k < KC; k += 4) {
                v2f b;
                b.x = B[(size_t)(k0 + k + koff + 0) * OUT_FEAT];
                b.y = B[(size_t)(k0 + k + koff + 1) * OUT_FEAT];
#pragma unroll
                for (int mt = 0; mt < MT_PER_BLK; ++mt) {
                    const float* ap =
                        &As[(mt * 16 + (lane & 15)) * LDS_STRIDE + k + koff];
                    const float2 af = *(const float2*)ap;   // ds_load_b64
                    v2f a; a.x = af.x; a.y = af.y;
                    acc[mt] = __builtin_amdgcn_wmma_f32_16x16x4_f32(
                        /*neg_a=*/false, a, /*neg_b=*/false, b,
                        /*c_mod=*/(short)0, acc[mt],
                        /*reuse_a=*/false, /*reuse_b=*/false);
                }
            }
        }
    }

    const float bn = bias[ncol];
#pragma unroll
    for (int mt = 0; mt < MT_PER_BLK; ++mt) {
        const int rbase = mbase + mt * 16 + (lane >> 4) * 8;  // D: VGPR r -> row
#pragma unroll
        for (int r = 0; r < 8; ++r) {
            const int row = rbase + r;
            if (row < N_NODES)
                out[(size_t)row * OUT_FEAT + ncol] = acc[mt][r] + bn;
        }
    }
}

// ---------------------------------------------------------------------------
extern "C" void kernel_launch(void* const* d_in, const int* in_sizes, int n_in,
                              void* d_out, int out_size, void* d_ws, size_t ws_size,
                              hipStream_t stream) {
    const float* feat   = (const float*)d_in[0];
    const float* coeff  = (const float*)d_in[1];
    const float* W      = (const float*)d_in[2];
    const float* h_bias = (const float*)d_in[3];
    const float* loopw  = (const float*)d_in[4];
    const int*   esrc   = (const int*)d_in[5];
    const int*   edst   = (const int*)d_in[6];
    const int*   etyp   = (const int*)d_in[7];
    float*       out    = (float*)d_out;

    // ws layout: agg [N_NODES][2*IN_FEAT] f32, then Wcomb [384][128] f32
    float* agg = (float*)d_ws;
    float* wc  = agg + (size_t)N_NODES * 2 * IN_FEAT;

    const int aggN4 = N_NODES * 2 * IN_FEAT / 4;           // 6.4M float4
    rgcn_zero_ws<<<(aggN4 + 255) / 256, 256, 0, stream>>>((float4*)agg, aggN4);

    const int wcN = (NUM_BASES + 1) * IN_FEAT * OUT_FEAT;  // 49152
    rgcn_build_wc<<<(wcN + 255) / 256, 256, 0, stream>>>(W, loopw, wc);

    const int nwaves  = (N_EDGES + EDGES_PER_WAVE - 1) / EDGES_PER_WAVE;  // 6250
    const int nblocks = (nwaves + 7) / 8;
    rgcn_edge_agg<<<nblocks, 256, 0, stream>>>(feat, coeff, esrc, edst, etyp, agg);

    const int mblocks = (N_NODES + 127) / 128;             // 782
    rgcn_gemm<<<mblocks, 256, 0, stream>>>(feat, agg, wc, h_bias, out);
}